// EffCapLayer_41918880809141
// MI455X (gfx1250) — compile-verified
//
#include <hip/hip_runtime.h>
#include <hip/hip_bf16.h>

// Problem constants
#define NB     4      // batch
#define HW_IN  32
#define HP     30     // output spatial (VALID 3x3)
#define BCAPS  32
#define CCAPS  32
#define DDIM   16     // P*P
#define SITES  (NB*HP*HP)          // 3600
#define UPOS   (NB*BCAPS*HP*HP)    // 115200 u positions
#define EPSV   1e-8f

typedef float v2f __attribute__((ext_vector_type(2)));
typedef float v8f __attribute__((ext_vector_type(8)));

// ---------------------------------------------------------------------------
// Kernel 1: depthwise 3D conv -> pose matrices u[n][b][h][w][i][j]
// u[n,b,h,w,i,j] = sum_{dd,kh,kw} x[n,h+kh,w+kw,b,j*4+dd] * dwk[b*4+i, dd, kh, kw]
// ---------------------------------------------------------------------------
__global__ __launch_bounds__(256) void u_conv_kernel(
    const float* __restrict__ x,     // [N][32][32][32][16]
    const float* __restrict__ dwk,   // [128][1][4][3][3]
    float* __restrict__ u_ws)        // [N][32][30][30][16]
{
  int tid = blockIdx.x * blockDim.x + threadIdx.x;
  if (tid >= UPOS) return;
  int n  = tid / (BCAPS * HP * HP);
  int r  = tid % (BCAPS * HP * HP);
  int b  = r / (HP * HP);
  int r2 = r % (HP * HP);
  int h  = r2 / HP;
  int w  = r2 % HP;

  float acc[16];
#pragma unroll
  for (int t = 0; t < 16; ++t) acc[t] = 0.f;

  for (int kh = 0; kh < 3; ++kh) {
    for (int kw = 0; kw < 3; ++kw) {
      const float* xp = x + ((((size_t)n * HW_IN + (h + kh)) * HW_IN + (w + kw)) * BCAPS + b) * DDIM;
      float xv[16];
#pragma unroll
      for (int t = 0; t < 16; ++t) xv[t] = xp[t];
      const float* kp = dwk + (size_t)b * 4 * 36 + kh * 3 + kw;
#pragma unroll
      for (int i = 0; i < 4; ++i) {
#pragma unroll
        for (int dd = 0; dd < 4; ++dd) {
          float kval = kp[i * 36 + dd * 9];
#pragma unroll
          for (int j = 0; j < 4; ++j)
            acc[i * 4 + j] += xv[j * 4 + dd] * kval;
        }
      }
    }
  }
  float* up = u_ws + (size_t)tid * 16;
#pragma unroll
  for (int t = 0; t < 16; ++t) up[t] = acc[t];
}

// ---------------------------------------------------------------------------
// Kernel 2: depthwise 2D conv + ReLU -> a_in[n][h][w][b]
// ---------------------------------------------------------------------------
__global__ __launch_bounds__(256) void a_conv_kernel(
    const float* __restrict__ a,     // [N][32][32][32][1]
    const float* __restrict__ adwk,  // [32][1][3][3]
    float* __restrict__ a_ws)        // [N][30][30][32]
{
  int tid = blockIdx.x * blockDim.x + threadIdx.x;
  if (tid >= SITES * BCAPS) return;
  int n  = tid / (HP * HP * BCAPS);
  int r  = tid % (HP * HP * BCAPS);
  int h  = r / (HP * BCAPS);
  int r2 = r % (HP * BCAPS);
  int w  = r2 / BCAPS;
  int b  = r2 % BCAPS;

  float acc = 0.f;
#pragma unroll
  for (int kh = 0; kh < 3; ++kh)
#pragma unroll
    for (int kw = 0; kw < 3; ++kw)
      acc += a[(((size_t)n * HW_IN + (h + kh)) * HW_IN + (w + kw)) * BCAPS + b] *
             adwk[b * 9 + kh * 3 + kw];
  a_ws[tid] = fmaxf(acc, 0.f);
}

// ---------------------------------------------------------------------------
// Kernel 3: fused votes (WMMA f32 16x16x4) + dynamic routing, one WG per site.
// The reference's v.reshape is a raw reinterpretation: site (n,h,w) consumes
// the contiguous F-range [F0, F0+32), F0 = h*960 + w*32, of the (n,b,h,w,c,i,k)
// vote array; b_o = F/900 is constant within each 4-aligned b-group.
// ---------------------------------------------------------------------------
__global__ __launch_bounds__(256) void capsule_route_kernel(
    const float* __restrict__ u_ws,    // [N][28800][16]  (b_o*900 + h_o*30 + w_o ordering)
    const float* __restrict__ a_ws,    // [N][30][30][32]
    const float* __restrict__ weight,  // [32][32][4][4]
    float* __restrict__ out_p,         // [site][32][16]
    float* __restrict__ out_a,         // [site][32]
    const int* __restrict__ riters)
{
  extern __shared__ float smem[];
  float* v_lds = smem;             // 16384 : votes v[b][c][d]
  float* coef  = v_lds + 16384;    // 1024  : routing coefficients [b][c]
  float* blog  = coef + 1024;      // 1024  : routing logits [b][c]
  float* sp    = blog + 1024;      // 512   : s then p [c][d]
  float* abv   = sp + 512;         // 32    : a_b[b]

  const int site = blockIdx.x;           // n*900 + h*30 + w
  const int n   = site / (HP * HP);
  const int rem = site % (HP * HP);
  const int h   = rem / HP;
  const int w   = rem % HP;
  const int F0  = h * 960 + w * 32;

  const int tid  = threadIdx.x;
  const int wv   = tid >> 5;             // wave id 0..7 -> b-tile
  const int lane = tid & 31;

  // ---- votes: D(16x16) = A(16x4: rows=(b,i), K=j) x B(4x16: cols=(c,k)) ----
  const float* ubase = u_ws + (size_t)n * (28800 * 16) + (size_t)F0 * 16;
  const int arow = lane & 15;            // m = b_loc*4 + i
  const int jsel = (lane >> 4) * 2;      // K element pair per lane half
  v2f A;
  {
    const float* up = ubase + wv * 64 + arow * 4 + jsel;
    A.x = up[0];
    A.y = up[1];
  }
  const int b_o   = (F0 + wv * 4) / 900; // constant across this wave's 4 b's
  const int c_loc = (lane & 15) >> 2;
  const int kk    = lane & 3;
  const int half  = (lane >> 4) * 8;

#pragma unroll
  for (int ct = 0; ct < 8; ++ct) {
    const int c = ct * 4 + c_loc;
    const float* wp = weight + (((size_t)b_o * CCAPS + c) * 4 + jsel) * 4 + kk;
    v2f Bm;
    Bm.x = wp[0];
    Bm.y = wp[4];                        // next j -> +4 floats
    v8f acc = {0.f, 0.f, 0.f, 0.f, 0.f, 0.f, 0.f, 0.f};
    acc = __builtin_amdgcn_wmma_f32_16x16x4_f32(
        /*neg_a=*/false, A, /*neg_b=*/false, Bm,
        /*c_mod=*/(short)0, acc, /*reuse_a=*/false, /*reuse_b=*/false);
#pragma unroll
    for (int r = 0; r < 8; ++r) {
      const int m    = r + half;         // row within tile
      const int bb   = wv * 4 + (m >> 2);
      const int ii   = m & 3;
      v_lds[bb * 512 + c * 16 + ii * 4 + kk] = acc[r];
    }
  }

  // init logits, load activations
  for (int idx = tid; idx < 1024; idx += 256) blog[idx] = 0.f;
  if (tid < 32) abv[tid] = a_ws[(size_t)site * 32 + tid];
  __syncthreads();

  const int iters = *riters;
  for (int it = 0; it < iters; ++it) {
    // coef[b][c] = softmax_c(blog[b,:]) * a_b[b]
    if (tid < 32) {
      const int bb = tid;
      float mx = blog[bb * 32];
      for (int c = 1; c < 32; ++c) mx = fmaxf(mx, blog[bb * 32 + c]);
      float ssum = 0.f;
      for (int c = 0; c < 32; ++c) {
        float ev = __expf(blog[bb * 32 + c] - mx);
        coef[bb * 32 + c] = ev;
        ssum += ev;
      }
      const float sc = abv[bb] / ssum;
      for (int c = 0; c < 32; ++c) coef[bb * 32 + c] *= sc;
    }
    __syncthreads();

    // s[c][d] = sum_b coef[b][c] * v[b][c][d]
    for (int cd = tid; cd < 512; cd += 256) {
      const int c = cd >> 4, d = cd & 15;
      float acc = 0.f;
#pragma unroll 4
      for (int bb = 0; bb < 32; ++bb)
        acc += coef[bb * 32 + c] * v_lds[bb * 512 + c * 16 + d];
      sp[cd] = acc;
    }
    __syncthreads();

    // squash in place: p = sq/(1+sq) * s / sqrt(sq+eps)
    if (tid < 32) {
      const int c = tid;
      float sq = 0.f;
#pragma unroll
      for (int d = 0; d < 16; ++d) { float t = sp[c * 16 + d]; sq += t * t; }
      const float scale = sq / (1.f + sq) / sqrtf(sq + EPSV);
#pragma unroll
      for (int d = 0; d < 16; ++d) sp[c * 16 + d] *= scale;
    }
    __syncthreads();

    if (it < iters - 1) {
      // blog[b][c] += sum_d v[b][c][d] * p[c][d]
      for (int bc = tid; bc < 1024; bc += 256) {
        const int bb = bc >> 5, c = bc & 31;
        float acc = 0.f;
#pragma unroll
        for (int d = 0; d < 16; ++d)
          acc += v_lds[bb * 512 + c * 16 + d] * sp[c * 16 + d];
        blog[bc] += acc;
      }
      __syncthreads();
    }
  }

  // outputs: p [site][32][16], a_out [site][32]
  for (int cd = tid; cd < 512; cd += 256)
    out_p[(size_t)site * 512 + cd] = sp[cd];
  if (tid < 32) {
    float sq = 0.f;
#pragma unroll
    for (int d = 0; d < 16; ++d) { float t = sp[tid * 16 + d]; sq += t * t; }
    out_a[(size_t)site * 32 + tid] = sqrtf(sq + EPSV);
  }
}

// ---------------------------------------------------------------------------
extern "C" void kernel_launch(void* const* d_in, const int* in_sizes, int n_in,
                              void* d_out, int out_size, void* d_ws, size_t ws_size,
                              hipStream_t stream) {
  (void)in_sizes; (void)n_in; (void)out_size; (void)ws_size;
  const float* x      = (const float*)d_in[0];
  const float* a      = (const float*)d_in[1];
  const float* dwk    = (const float*)d_in[2];
  const float* adwk   = (const float*)d_in[3];
  const float* wgt    = (const float*)d_in[4];
  const int*   riters = (const int*)d_in[5];

  float* u_ws = (float*)d_ws;                     // 1,843,200 floats
  float* a_ws = u_ws + (size_t)UPOS * 16;         //   115,200 floats

  float* out_p = (float*)d_out;                   // 1,843,200 floats
  float* out_a = out_p + (size_t)SITES * 512;     //   115,200 floats

  u_conv_kernel<<<(UPOS + 255) / 256, 256, 0, stream>>>(x, dwk, u_ws);
  a_conv_kernel<<<(SITES * BCAPS + 255) / 256, 256, 0, stream>>>(a, adwk, a_ws);

  const size_t shmem = (16384 + 1024 + 1024 + 512 + 32) * sizeof(float);
  capsule_route_kernel<<<SITES, 256, shmem, stream>>>(u_ws, a_ws, wgt, out_p, out_a, riters);
}